// Interaction_35777077575986
// MI455X (gfx1250) — compile-verified
//
#include <hip/hip_runtime.h>
#include <math.h>

// ---------------------------------------------------------------------------
// MACE-like interaction block for MI455X (gfx1250, wave32, WMMA).
//   K1: linear_up  h0 = nf0@Wup0, h1[:,:,i] = nf1[:,:,i]@Wup1   (fp32 WMMA)
//   K2: fused radial MLP (fp32 WMMA) + edge gather + atomic scatter to T0/T1
//   K3: 1/16 scale, Q projection, per-species residual, concat output
// All hot-path indexing is 32-bit (max buffer 38.4M elements) to avoid
// v_mul_u64 / v_add_nc_u64 chains seen in the previous round's codegen.
// ---------------------------------------------------------------------------

typedef float v2f __attribute__((ext_vector_type(2)));
typedef float v8f __attribute__((ext_vector_type(8)));

#define AVG_INV (1.0f / 16.0f)

// D = A(16x4) * B(4x16) + C, all fp32, wave32.
__device__ __forceinline__ v8f wmma4(v2f a, v2f b, v8f c) {
  return __builtin_amdgcn_wmma_f32_16x16x4_f32(
      /*neg_a=*/false, a, /*neg_b=*/false, b,
      /*c_mod=*/(short)0, c, /*reuse_a=*/false, /*reuse_b=*/false);
}

// Hardware f32 atomic add (global_atomic_add_f32), relaxed, device scope.
__device__ __forceinline__ void atomic_add_f32(float* p, float v) {
  __hip_atomic_fetch_add(p, v, __ATOMIC_RELAXED, __HIP_MEMORY_SCOPE_AGENT);
}

// ---------------------------------------------------------------------------
// K0: zero the T0/T1 accumulators (workspace is poisoned between calls).
// ---------------------------------------------------------------------------
__global__ void zero_f32(float* __restrict__ p, int n) {
  int i = blockIdx.x * blockDim.x + threadIdx.x;
  const int stride = gridDim.x * blockDim.x;
  for (; i < n; i += stride) p[i] = 0.0f;
}

// ---------------------------------------------------------------------------
// K1: linear_up. One block handles a 16-node row tile. 8 waves cover the
// 16 (matrix, col-tile) pairs: mat 0 -> h0, mats 1..3 -> h1 components.
// ES is the element stride (1 for h0 path, 3 for h1 components), compile-time
// so index math folds to shifts/adds and load-offset immediates.
// ---------------------------------------------------------------------------
template <int ES>
__device__ __forceinline__ void lin_tile(const float* __restrict__ X, int eo,
                                         const float* __restrict__ W,
                                         float* __restrict__ O, int row0,
                                         int col, int half, int l16) {
  v8f acc = {0.f, 0.f, 0.f, 0.f, 0.f, 0.f, 0.f, 0.f};
  const int arow = (row0 + l16) * 64;
#pragma unroll
  for (int ks = 0; ks < 16; ++ks) {  // K = 64, 4 per WMMA
    const int k = 4 * ks + 2 * half;
    v2f a, b;
    a.x = X[(arow + k) * ES + eo];
    a.y = X[(arow + k + 1) * ES + eo];
    b.x = W[k * 64 + col];
    b.y = W[(k + 1) * 64 + col];
    acc = wmma4(a, b, acc);
  }
#pragma unroll
  for (int r = 0; r < 8; ++r) {
    const int row = row0 + r + 8 * half;
    O[(row * 64 + col) * ES + eo] = acc[r];
  }
}

__global__ void __launch_bounds__(256)
linear_up_kernel(const float* __restrict__ nf0,   // [N,64]
                 const float* __restrict__ nf1,   // [N,64,3]
                 const float* __restrict__ Wup0,  // [64,64]
                 const float* __restrict__ Wup1,  // [64,64]
                 float* __restrict__ h0,          // [N,64]
                 float* __restrict__ h1) {        // [N,64,3]
  const int lane = threadIdx.x & 31;
  const int wv   = threadIdx.x >> 5;
  const int half = lane >> 4;
  const int l16  = lane & 15;
  const int row0 = blockIdx.x * 16;

#pragma unroll
  for (int t = 0; t < 2; ++t) {
    const int pair = wv + 8 * t;        // 0..15
    const int mat  = pair >> 2;         // 0..3
    const int col  = (pair & 3) * 16 + l16;
    if (mat == 0)
      lin_tile<1>(nf0, 0, Wup0, h0, row0, col, half, l16);
    else
      lin_tile<3>(nf1, mat - 1, Wup1, h1, row0, col, half, l16);
  }
}

// ---------------------------------------------------------------------------
// K2: fused radial MLP + edge message + scatter.
// Block = 256 threads (8 waves), 64 edges per block in 4 chunks of 16.
// W2 WMMA fragments stay in registers across chunks (64 VGPRs/lane).
// ---------------------------------------------------------------------------
#define HST 72    // LDS row stride for hidden[16][64]  (bank-conflict pad)
#define WST 260   // LDS row stride for w[16][256]

__global__ void __launch_bounds__(256)
edge_kernel(const float* __restrict__ ef,    // [E,8]
            const float* __restrict__ sh,    // [E,3] edge_attrs1
            const float* __restrict__ cut,   // [E]
            const float* __restrict__ W1,    // [8,64]
            const float* __restrict__ W2,    // [64,256]
            const int*   __restrict__ ei,    // [2,E]
            const float* __restrict__ h0,    // [N,64]
            const float* __restrict__ h1,    // [N,64,3]
            float* __restrict__ T0,          // [N,64]
            float* __restrict__ T1,          // [N,64,3]
            int E) {
  __shared__ float hid[16 * HST];
  __shared__ float wls[16 * WST];
  __shared__ int   esrc[16], edst[16];
  __shared__ float eshs[16][3];
  __shared__ float ecut[16];

  const int tid  = threadIdx.x;
  const int lane = tid & 31;
  const int wv   = tid >> 5;
  const int half = lane >> 4;
  const int l16  = lane & 15;

  // Preload W2 B-fragments: wave wv owns output col tiles wv and wv+8.
  v2f bw2[2][16];
#pragma unroll
  for (int t = 0; t < 2; ++t) {
    const int col = (wv + 8 * t) * 16 + l16;
#pragma unroll
    for (int ks = 0; ks < 16; ++ks) {
      const int k = 4 * ks + 2 * half;
      bw2[t][ks].x = W2[k * 256 + col];
      bw2[t][ks].y = W2[(k + 1) * 256 + col];
    }
  }
  // Preload W1 B-fragments (waves 0..3 compute the 16x64 hidden tile).
  v2f bw1[2];
  if (wv < 4) {
    const int col = wv * 16 + l16;
#pragma unroll
    for (int ks = 0; ks < 2; ++ks) {
      const int k = 4 * ks + 2 * half;
      bw1[ks].x = W1[k * 64 + col];
      bw1[ks].y = W1[(k + 1) * 64 + col];
    }
  }

  for (int chunk = 0; chunk < 4; ++chunk) {
    const int e0 = blockIdx.x * 64 + chunk * 16;

    // Prefetch next chunk's streamed edge data into cache (gfx1250
    // global_prefetch path) while this chunk computes.
    if (chunk < 3 && tid < 16) {
      const int en = e0 + 16 + tid;
      __builtin_prefetch(&ef[en * 8], 0, 3);
      __builtin_prefetch(&ei[en], 0, 3);
      __builtin_prefetch(&ei[E + en], 0, 3);
      __builtin_prefetch(&sh[en * 3], 0, 3);
    }

    if (tid < 16) {                      // edge metadata for this chunk
      const int e  = e0 + tid;
      esrc[tid]    = ei[e];
      edst[tid]    = ei[E + e];
      eshs[tid][0] = sh[e * 3 + 0];
      eshs[tid][1] = sh[e * 3 + 1];
      eshs[tid][2] = sh[e * 3 + 2];
      ecut[tid]    = cut[e];
    }

    // GEMM1: hidden = silu(ef[16,8] @ W1[8,64])
    if (wv < 4) {
      v8f acc = {0.f, 0.f, 0.f, 0.f, 0.f, 0.f, 0.f, 0.f};
#pragma unroll
      for (int ks = 0; ks < 2; ++ks) {   // K = 8
        const int k = 4 * ks + 2 * half;
        v2f a;
        a.x = ef[(e0 + l16) * 8 + k];
        a.y = ef[(e0 + l16) * 8 + k + 1];
        acc = wmma4(a, bw1[ks], acc);
      }
#pragma unroll
      for (int r = 0; r < 8; ++r) {
        const float x = acc[r];
        hid[(r + 8 * half) * HST + wv * 16 + l16] = x / (1.0f + __expf(-x));
      }
    }
    __syncthreads();

    // GEMM2: w[16,256] = hidden[16,64] @ W2[64,256]
    {
      v8f acc0 = {0.f, 0.f, 0.f, 0.f, 0.f, 0.f, 0.f, 0.f};
      v8f acc1 = {0.f, 0.f, 0.f, 0.f, 0.f, 0.f, 0.f, 0.f};
#pragma unroll
      for (int ks = 0; ks < 16; ++ks) {  // K = 64
        const int k = 4 * ks + 2 * half;
        v2f a;
        a.x  = hid[l16 * HST + k];
        a.y  = hid[l16 * HST + k + 1];
        acc0 = wmma4(a, bw2[0][ks], acc0);
        acc1 = wmma4(a, bw2[1][ks], acc1);
      }
#pragma unroll
      for (int r = 0; r < 8; ++r) {
        const int m = r + 8 * half;
        wls[m * WST + wv * 16 + l16]       = acc0[r];
        wls[m * WST + (wv + 8) * 16 + l16] = acc1[r];
      }
    }
    __syncthreads();

    // Message + scatter: 16 edges x 64 channels, 4 (e,c) pairs per thread.
    for (int p = tid; p < 16 * 64; p += 256) {
      const int e = p >> 6;
      const int c = p & 63;
      const int s = esrc[e];
      const int d = edst[e];
      const float w00 = wls[e * WST + c];
      const float w11 = wls[e * WST + 64 + c];
      const float w01 = wls[e * WST + 128 + c];
      const float w10 = wls[e * WST + 192 + c];
      const float ct  = ecut[e];
      const float sx = eshs[e][0], sy = eshs[e][1], sz = eshs[e][2];
      const float h0v = h0[s * 64 + c];
      const float* h1p = &h1[(s * 64 + c) * 3];
      const float hx = h1p[0], hy = h1p[1], hz = h1p[2];

      const float m0 = (w00 * h0v + w11 * (hx * sx + hy * sy + hz * sz)) * ct;
      atomic_add_f32(&T0[d * 64 + c], m0);

      const float b = w01 * h0v;
      float* t1p = &T1[(d * 64 + c) * 3];
      atomic_add_f32(t1p + 0, (b * sx + w10 * hx) * ct);
      atomic_add_f32(t1p + 1, (b * sy + w10 * hy) * ct);
      atomic_add_f32(t1p + 2, (b * sz + w10 * hz) * ct);
    }
    __syncthreads();   // protect LDS + metadata before next chunk
  }
}

// ---------------------------------------------------------------------------
// K3: per-node output: mi = (T/16) @ Q, residual via per-species mixing,
// concat [mi0 | mi1 | res0 | res1] -> out[n, 512].
// ---------------------------------------------------------------------------
__global__ void __launch_bounds__(256)
out_kernel(const float* __restrict__ T0,     // [N,64]
           const float* __restrict__ T1,     // [N,64,3]
           const float* __restrict__ attrs,  // [N,S] one-hot
           const float* __restrict__ Wsc0,   // [S,64,64]
           const float* __restrict__ Wsc1,   // [S,64,64]
           const float* __restrict__ Q0,     // [1,1]
           const float* __restrict__ Q1,     // [3,3]
           float* __restrict__ out,          // [N,512]
           int S) {
  const int n   = blockIdx.x;
  const int tid = threadIdx.x;
  __shared__ float mi[256];   // [0:64) = mi0, [64:256) = mi1 in (c,i) order
  __shared__ float q1s[9];

  if (tid < 9) q1s[tid] = Q1[tid];
  __syncthreads();

  if (tid < 64) {
    mi[tid] = T0[n * 64 + tid] * AVG_INV * Q0[0];
  } else {
    const int p = tid - 64;
    const int c = p / 3, i = p % 3;
    const float* tp = &T1[(n * 64 + c) * 3];
    mi[tid] = (tp[0] * q1s[0 * 3 + i] + tp[1] * q1s[1 * 3 + i] +
               tp[2] * q1s[2 * 3 + i]) * AVG_INV;
  }
  __syncthreads();

  float* orow = out + n * 512;
  orow[tid] = mi[tid];        // cols [0,256): mi0 | mi1

  float acc = 0.0f;
  if (tid < 64) {
    const int o = tid;
    for (int s = 0; s < S; ++s) {
      const float a = attrs[n * S + s];
      if (a != 0.0f) {
        const float* Wp = &Wsc0[s * 64 * 64 + o];
        float dsum = 0.0f;
#pragma unroll 8
        for (int c = 0; c < 64; ++c) dsum += mi[c] * Wp[c * 64];
        acc += a * dsum;
      }
    }
    orow[256 + o] = acc;      // cols [256,320): res0
  } else {
    const int p = tid - 64;
    const int o = p / 3, i = p % 3;
    for (int s = 0; s < S; ++s) {
      const float a = attrs[n * S + s];
      if (a != 0.0f) {
        const float* Wp = &Wsc1[s * 64 * 64 + o];
        float dsum = 0.0f;
#pragma unroll 8
        for (int c = 0; c < 64; ++c) dsum += mi[64 + c * 3 + i] * Wp[c * 64];
        acc += a * dsum;
      }
    }
    orow[320 + p] = acc;      // cols [320,512): res1
  }
}

// ---------------------------------------------------------------------------
extern "C" void kernel_launch(void* const* d_in, const int* in_sizes, int n_in,
                              void* d_out, int out_size, void* d_ws,
                              size_t ws_size, hipStream_t stream) {
  const float* nf0   = (const float*)d_in[0];
  const float* nf1   = (const float*)d_in[1];
  const float* attrs = (const float*)d_in[2];
  const float* ef    = (const float*)d_in[3];
  const float* sh1   = (const float*)d_in[4];
  const float* cut   = (const float*)d_in[5];
  const float* Wup0  = (const float*)d_in[6];
  const float* Wup1  = (const float*)d_in[7];
  const float* W1    = (const float*)d_in[8];
  const float* W2    = (const float*)d_in[9];
  const float* Wsc0  = (const float*)d_in[10];
  const float* Wsc1  = (const float*)d_in[11];
  const float* Q0    = (const float*)d_in[12];
  const float* Q1    = (const float*)d_in[13];
  const int*   ei    = (const int*)d_in[14];

  const int N = in_sizes[0] / 64;    // 50000 (multiple of 16)
  const int E = in_sizes[5];         // 800000 (multiple of 64)
  const int S = in_sizes[2] / N;     // 10

  // Workspace layout: h0 [N,64] | h1 [N,64,3] | T0 [N,64] | T1 [N,64,3]
  float* h0 = (float*)d_ws;
  float* h1 = h0 + (size_t)N * 64;
  float* T0 = h1 + (size_t)N * 192;
  float* T1 = T0 + (size_t)N * 64;

  zero_f32<<<2048, 256, 0, stream>>>(T0, N * 256);
  linear_up_kernel<<<N / 16, 256, 0, stream>>>(nf0, nf1, Wup0, Wup1, h0, h1);
  edge_kernel<<<E / 64, 256, 0, stream>>>(ef, sh1, cut, W1, W2, ei, h0, h1,
                                          T0, T1, E);
  out_kernel<<<N, 256, 0, stream>>>(T0, T1, attrs, Wsc0, Wsc1, Q0, Q1,
                                    (float*)d_out, S);
}